// TransformerBlock_81458349736270
// MI455X (gfx1250) — compile-verified
//
#include <hip/hip_runtime.h>
#include <math.h>

// ---------------------------------------------------------------------------
// TransformerBlock for MI455X (gfx1250, wave32, WMMA f32 16x16x4,
// async global->LDS staging with double buffering).
//   B=8, T=1024, D=768, H=12, HD=64, DFF=3072
// ---------------------------------------------------------------------------

constexpr int Bn   = 8;
constexpr int Tn   = 1024;
constexpr int Dn   = 768;
constexpr int Hn   = 12;
constexpr int HDn  = 64;
constexpr int DFFn = 3072;

// LDS strides (floats): multiples of 4 (16B async writes) and bank-conflict
// free for the WMMA fragment read patterns.
constexpr int LDA  = 36;     // A tile row stride (64 rows x 32 cols)
constexpr int LDB  = 72;     // B tile row stride (32 rows x 64 cols)
constexpr int LDQ  = 68;     // Q tile row stride (16 rows x 64 cols)
constexpr int SST  = 1028;   // score-buffer row stride (16 rows x 1024 cols)

typedef float v2f __attribute__((ext_vector_type(2)));
typedef float v8f __attribute__((ext_vector_type(8)));

#define WMMA_F32(a, b, c) \
  __builtin_amdgcn_wmma_f32_16x16x4_f32(false, (a), false, (b), (short)0, (c), false, false)

// Async global->LDS copy, 16 bytes per lane, tracked by ASYNCcnt.
// LDS operand is the wave-relative LDS offset == low 32 bits of the flat
// shared-pointer address (LDS aperture base has zero low bits).
__device__ __forceinline__ void async_copy_b128(void* lds_dst, const void* gsrc) {
  unsigned lds = (unsigned)(uintptr_t)lds_dst;
  unsigned long long ga = (unsigned long long)(uintptr_t)gsrc;
  asm volatile("global_load_async_to_lds_b128 %0, %1, off"
               :: "v"(lds), "v"(ga)
               : "memory");
}
#define WAIT_ASYNCCNT(n) asm volatile("s_wait_asynccnt " #n ::: "memory")

__device__ __forceinline__ float gelu_exact(float x) {
  return 0.5f * x * (1.0f + erff(x * 0.70710678118654752440f));
}

__device__ __forceinline__ float block_sum_256(float v, float* sbuf) {
  int t = threadIdx.x;
  sbuf[t] = v; __syncthreads();
  #pragma unroll
  for (int s = 128; s > 0; s >>= 1) {
    if (t < s) sbuf[t] += sbuf[t + s];
    __syncthreads();
  }
  float r = sbuf[0];
  __syncthreads();
  return r;
}

// ---------------------------------------------------------------------------
// LayerNorm: one block (256 threads) per token row; D = 768 = 3 * 256.
// ---------------------------------------------------------------------------
__global__ __launch_bounds__(256) void layernorm_kernel(
    const float* __restrict__ x, const float* __restrict__ g,
    const float* __restrict__ b, float* __restrict__ y) {
  __shared__ float sbuf[256];
  const size_t row = blockIdx.x;
  const float* xr = x + row * Dn;
  float*       yr = y + row * Dn;
  float v[3];
  float s = 0.f;
  #pragma unroll
  for (int i = 0; i < 3; ++i) { v[i] = xr[threadIdx.x + i * 256]; s += v[i]; }
  s = block_sum_256(s, sbuf);
  const float mu = s * (1.0f / Dn);
  float sq = 0.f;
  #pragma unroll
  for (int i = 0; i < 3; ++i) { float d = v[i] - mu; sq += d * d; }
  sq = block_sum_256(sq, sbuf);
  const float inv = rsqrtf(sq * (1.0f / Dn) + 1e-5f);
  #pragma unroll
  for (int i = 0; i < 3; ++i) {
    int c = threadIdx.x + i * 256;
    yr[c] = (v[i] - mu) * inv * g[c] + b[c];
  }
}

// ---------------------------------------------------------------------------
// Generic GEMM: C[M,N] = act(A[M,K] @ W[K,N] + bias[N]) (+ R[M,N])
// Block = 256 threads (8 waves), block tile 64x64, K in 32-chunks.
// Double-buffered LDS tiles filled by GLOBAL_LOAD_ASYNC_TO_LDS_B128:
// while WMMAs consume chunk k, chunk k+1 streams into the other buffer
// (only the 4 just-issued async ops per wave remain outstanding:
//  s_wait_asynccnt 4).
// Requires: M % 64 == 0, N % 64 == 0, K % 32 == 0 (true for all shapes here).
// ACT: 0 = none, 1 = exact GELU.
// ---------------------------------------------------------------------------
template <int ACT>
__global__ __launch_bounds__(256) void gemm_bias_kernel(
    const float* __restrict__ A, const float* __restrict__ W,
    const float* __restrict__ bias, const float* __restrict__ R,
    float* __restrict__ C, int M, int N, int K) {
  __shared__ float ldsA[2][64 * LDA];
  __shared__ float ldsB[2][32 * LDB];
  const int lane = threadIdx.x & 31;
  const int wave = threadIdx.x >> 5;
  const int wm = wave & 3, wn = wave >> 2;
  const int ml = lane & 15;
  const int kh = (lane >> 4) << 1;          // 0 or 2 (K-pair select per ISA layout)
  const int mblk = blockIdx.y * 64;
  const int nblk = blockIdx.x * 64;

  v8f acc0 = {}; v8f acc1 = {};

  // Per-thread async-copy assignments (2 x 16B for A, 2 x 16B for B).
  // A tile: 64x32 floats = 512 chunks;  B tile: 32x64 floats = 512 chunks.
  const int ca0 = threadIdx.x, ca1 = threadIdx.x + 256;

  auto issue_stage = [&](int buf, int k0) {
    float* la = &ldsA[buf][0];
    float* lb = &ldsB[buf][0];
    {
      const int r = ca0 >> 3, c = (ca0 & 7) << 2;
      async_copy_b128(&la[r * LDA + c], &A[(size_t)(mblk + r) * K + k0 + c]);
    }
    {
      const int r = ca1 >> 3, c = (ca1 & 7) << 2;
      async_copy_b128(&la[r * LDA + c], &A[(size_t)(mblk + r) * K + k0 + c]);
    }
    {
      const int r = ca0 >> 4, c = (ca0 & 15) << 2;
      async_copy_b128(&lb[r * LDB + c], &W[(size_t)(k0 + r) * N + nblk + c]);
    }
    {
      const int r = ca1 >> 4, c = (ca1 & 15) << 2;
      async_copy_b128(&lb[r * LDB + c], &W[(size_t)(k0 + r) * N + nblk + c]);
    }
  };

  const int nch = K >> 5;                   // number of 32-wide K chunks
  issue_stage(0, 0);
  for (int ch = 0; ch < nch; ++ch) {
    const int cur = ch & 1;
    if (ch + 1 < nch) {
      issue_stage(cur ^ 1, (ch + 1) << 5);  // prefetch next chunk
      WAIT_ASYNCCNT(4);                     // current chunk's 4 ops complete
    } else {
      WAIT_ASYNCCNT(0);
    }
    __syncthreads();                        // publish LDS tile to all waves

    const float* pa = &ldsA[cur][(wm * 16 + ml) * LDA];
    const float* pb = &ldsB[cur][wn * 32 + ml];
    #pragma unroll
    for (int kk = 0; kk < 32; kk += 4) {
      v2f a;  a.x  = pa[kk + kh];                 a.y  = pa[kk + kh + 1];
      v2f b0; b0.x = pb[(kk + kh) * LDB];         b0.y = pb[(kk + kh + 1) * LDB];
      v2f b1; b1.x = pb[(kk + kh) * LDB + 16];    b1.y = pb[(kk + kh + 1) * LDB + 16];
      acc0 = WMMA_F32(a, b0, acc0);
      acc1 = WMMA_F32(a, b1, acc1);
    }
    __syncthreads();                        // buffer free before re-issue
  }

  // Epilogue per ISA C/D layout: VGPR v -> row v (+8 for upper lane half).
  const int rowbase = mblk + wm * 16 + ((lane >> 4) << 3);
  const int c0 = nblk + wn * 32 + ml;
  const float bi0 = bias[c0], bi1 = bias[c0 + 16];
  #pragma unroll
  for (int v = 0; v < 8; ++v) {
    const int row = rowbase + v;
    float v0 = acc0[v] + bi0;
    float v1 = acc1[v] + bi1;
    if (ACT == 1) { v0 = gelu_exact(v0); v1 = gelu_exact(v1); }
    const size_t o = (size_t)row * N + c0;
    if (R) { v0 += R[o]; v1 += R[o + 16]; }
    C[o] = v0; C[o + 16] = v1;
  }
}

// ---------------------------------------------------------------------------
// Fused causal attention per (b, h, 16-query tile):
//   scores (WMMA, causal mask) -> softmax in LDS -> single HBM write of
//   attn_weights -> AV GEMM (WMMA) reading A-fragments straight from LDS.
// Block = 128 threads (4 waves). Dynamic LDS (~70 KB of 320 KB/WGP):
//   Q tile 16xLDQ (async-staged), score buffer 16xSST, reduction scratch.
// ---------------------------------------------------------------------------
__global__ __launch_bounds__(128) void attn_fused_kernel(
    const float* __restrict__ qkv,     // [B,T,3D]
    float* __restrict__ attnW,         // [B*H, T, T]
    float* __restrict__ ctx) {         // [B,T,D]
  extern __shared__ float smem[];
  float* Sq   = smem;                  // 16*LDQ
  float* Ss   = Sq + 16 * LDQ;         // 16*SST
  float* sred = Ss + 16 * SST;         // 16*9

  const int qtile = blockIdx.x;
  const int z     = blockIdx.y;        // b*H + h
  const int bb    = z / Hn;
  const int hh    = z % Hn;
  const int tid   = threadIdx.x;
  const int lane  = tid & 31;
  const int wave  = tid >> 5;
  const int ml    = lane & 15;
  const int kh    = (lane >> 4) << 1;
  const int q0    = qtile * 16;
  const float NEG_INF = -__builtin_inff();

  // Stage Q tile (16 x 64 floats = 256 x 16B chunks) via async copy.
  {
    const int c0i = tid, c1i = tid + 128;
    const int r0 = c0i >> 4, cc0 = (c0i & 15) << 2;
    const int r1 = c1i >> 4, cc1 = (c1i & 15) << 2;
    const float* qb = qkv + (size_t)(bb * Tn + q0) * (3 * Dn) + hh * HDn;
    async_copy_b128(&Sq[r0 * LDQ + cc0], qb + (size_t)r0 * (3 * Dn) + cc0);
    async_copy_b128(&Sq[r1 * LDQ + cc1], qb + (size_t)r1 * (3 * Dn) + cc1);
    WAIT_ASYNCCNT(0);
  }
  __syncthreads();

  // Scores: wave handles ktiles {wave, wave+4, ...}; branch is wave-uniform.
  for (int kt = wave; kt < Tn / 16; kt += 4) {
    const int k0 = kt * 16;
    const int rb = (lane >> 4) << 3;
    if (k0 <= q0 + 15) {
      v8f acc = {};
      const float* kb = qkv + (size_t)(bb * Tn + k0 + ml) * (3 * Dn) + Dn + hh * HDn;
      #pragma unroll
      for (int d = 0; d < HDn; d += 4) {
        v2f a;  a.x  = Sq[ml * LDQ + d + kh]; a.y = Sq[ml * LDQ + d + kh + 1];
        v2f bf; bf.x = kb[d + kh];            bf.y = kb[d + kh + 1];
        acc = WMMA_F32(a, bf, acc);
      }
      #pragma unroll
      for (int v = 0; v < 8; ++v) {
        const int qr = rb + v;               // local query row
        const int kc = k0 + ml;              // global key col
        float s = acc[v] * 0.125f;           // 1/sqrt(64)
        if (kc > q0 + qr) s = NEG_INF;       // causal mask
        Ss[qr * SST + kc] = s;
      }
    } else {
      #pragma unroll
      for (int v = 0; v < 8; ++v)
        Ss[(rb + v) * SST + k0 + ml] = NEG_INF;
    }
  }
  __syncthreads();

  // Softmax over each of the 16 rows (8 threads per row).
  const int r  = tid >> 3;
  const int c0 = tid & 7;
  float m = NEG_INF;
  for (int c = c0; c < Tn; c += 8) m = fmaxf(m, Ss[r * SST + c]);
  sred[r * 9 + c0] = m;
  __syncthreads();
  float rm = NEG_INF;
  #pragma unroll
  for (int i = 0; i < 8; ++i) rm = fmaxf(rm, sred[r * 9 + i]);
  float ssum = 0.f;
  for (int c = c0; c < Tn; c += 8) {
    float e = __expf(Ss[r * SST + c] - rm);  // exp(-inf - m) = 0 above diagonal
    Ss[r * SST + c] = e;
    ssum += e;
  }
  __syncthreads();
  sred[r * 9 + c0] = ssum;
  __syncthreads();
  float rs = 0.f;
  #pragma unroll
  for (int i = 0; i < 8; ++i) rs += sred[r * 9 + i];
  const float inv = 1.f / rs;
  for (int c = c0; c < Tn; c += 8) Ss[r * SST + c] *= inv;
  __syncthreads();

  // Single coalesced write of attention weights.
  const size_t wbase = ((size_t)z * Tn + q0) * Tn;
  for (int i = tid; i < 16 * Tn; i += 128) {
    int rr = i >> 10, cc = i & (Tn - 1);
    attnW[wbase + (size_t)rr * Tn + cc] = Ss[rr * SST + cc];
  }

  // Context GEMM: wave -> 16-wide column tile of HD; causal K truncation.
  const int kmax = q0 + 16;                  // weights are exactly 0 beyond this
  v8f acc = {};
  const float* vb = qkv + (size_t)(bb * Tn) * (3 * Dn) + 2 * Dn + hh * HDn + wave * 16;
  for (int k = 0; k < kmax; k += 4) {
    v2f a;  a.x  = Ss[ml * SST + k + kh];    a.y  = Ss[ml * SST + k + kh + 1];
    v2f bf; bf.x = vb[(size_t)(k + kh) * (3 * Dn) + ml];
    bf.y = vb[(size_t)(k + kh + 1) * (3 * Dn) + ml];
    acc = WMMA_F32(a, bf, acc);
  }
  const int col = wave * 16 + ml;
  const int rb  = (lane >> 4) << 3;
  #pragma unroll
  for (int v = 0; v < 8; ++v) {
    const int qr = q0 + rb + v;
    ctx[(size_t)(bb * Tn + qr) * Dn + hh * HDn + col] = acc[v];
  }
}

// ---------------------------------------------------------------------------
// Host launch
// ---------------------------------------------------------------------------
extern "C" void kernel_launch(void* const* d_in, const int* in_sizes, int n_in,
                              void* d_out, int out_size, void* d_ws, size_t ws_size,
                              hipStream_t stream) {
  (void)in_sizes; (void)n_in; (void)out_size; (void)ws_size;
  const float* x     = (const float*)d_in[0];
  const float* ln1_g = (const float*)d_in[1];
  const float* ln1_b = (const float*)d_in[2];
  const float* w_qkv = (const float*)d_in[3];
  const float* b_qkv = (const float*)d_in[4];
  const float* w_out = (const float*)d_in[5];
  const float* b_out = (const float*)d_in[6];
  const float* ln2_g = (const float*)d_in[7];
  const float* ln2_b = (const float*)d_in[8];
  const float* w_fc1 = (const float*)d_in[9];
  const float* b_fc1 = (const float*)d_in[10];
  const float* w_fc2 = (const float*)d_in[11];
  const float* b_fc2 = (const float*)d_in[12];

  float* outx  = (float*)d_out;                         // [B,T,D]
  float* attnW = outx + (size_t)Bn * Tn * Dn;           // [B,H,T,T]

  const size_t nTok = (size_t)Bn * Tn;                  // 8192
  float* ws  = (float*)d_ws;
  float* h   = ws;                                      // nTok*D   (ln1 / ln2 out)
  float* big = h + nTok * Dn;                           // nTok*DFF (qkv, then ff)
  float* ctx = big + nTok * DFFn;                       // nTok*D
  float* x1  = ctx + nTok * Dn;                         // nTok*D

  const int M = Bn * Tn;                                // 8192

  // 1) LN1
  layernorm_kernel<<<dim3((unsigned)nTok), 256, 0, stream>>>(x, ln1_g, ln1_b, h);
  // 2) QKV GEMM: [8192,768] x [768,2304]
  gemm_bias_kernel<0><<<dim3(3 * Dn / 64, M / 64), 256, 0, stream>>>(
      h, w_qkv, b_qkv, nullptr, big, M, 3 * Dn, Dn);
  // 3) Fused attention (scores + softmax + weights write + AV)
  {
    const size_t shbytes = (size_t)(16 * LDQ + 16 * SST + 16 * 9) * sizeof(float);
    attn_fused_kernel<<<dim3(Tn / 16, Bn * Hn), 128, shbytes, stream>>>(big, attnW, ctx);
  }
  // 4) Out projection + residual(x)
  gemm_bias_kernel<0><<<dim3(Dn / 64, M / 64), 256, 0, stream>>>(
      ctx, w_out, b_out, x, x1, M, Dn, Dn);
  // 5) LN2
  layernorm_kernel<<<dim3((unsigned)nTok), 256, 0, stream>>>(x1, ln2_g, ln2_b, h);
  // 6) FC1 + exact GELU
  gemm_bias_kernel<1><<<dim3(DFFn / 64, M / 64), 256, 0, stream>>>(
      h, w_fc1, b_fc1, nullptr, big, M, DFFn, Dn);
  // 7) FC2 + residual(x1) -> final output
  gemm_bias_kernel<0><<<dim3(Dn / 64, M / 64), 256, 0, stream>>>(
      big, w_fc2, b_fc2, x1, outx, M, Dn, DFFn);
}